// CTRNN_MD_47674136985612
// MI455X (gfx1250) — compile-verified
//
#include <hip/hip_runtime.h>

typedef __attribute__((ext_vector_type(2))) float v2f;
typedef __attribute__((ext_vector_type(8))) float v8f;

#define T_  256
#define B_  256
#define I_  512
#define H_  1024
#define MD_ 10
#define ALPHA_ 0.5f

// gates[b,h] = sum_m sub_id[b,m] * mask[m,h]
__global__ __launch_bounds__(256) void ctrnn_gates_kernel(
    const float* __restrict__ sub_id,   // B x MD
    const float* __restrict__ mask,     // MD x H
    float* __restrict__ gates)          // B x H
{
    int idx = blockIdx.x * 256 + threadIdx.x;   // over B*H
    int b = idx >> 10;          // / H_
    int h = idx & (H_ - 1);
    float s = 0.0f;
#pragma unroll
    for (int m = 0; m < MD_; ++m)
        s += sub_id[b * MD_ + m] * mask[m * H_ + h];
    gates[idx] = s;
}

// Fully-parallel input projection: ext = x @ W_in^T + b_in, over R = T*B rows.
// Each wave: 64x64 macro tile = 4x4 WMMA tiles (16 FLOP/B from L2).
// Block = 8 waves covering 128 rows x 256 cols. Grid = 512 x 4 = 2048 blocks.
__global__ __launch_bounds__(256) void ctrnn_ext_kernel(
    const float* __restrict__ x,     // (T*B) x I
    const float* __restrict__ W_in,  // H x I
    const float* __restrict__ b_in,  // H
    float* __restrict__ ext)         // (T*B) x H
{
    const int wave = threadIdx.x >> 5;     // 0..7
    const int lane = threadIdx.x & 31;
    const int br = blockIdx.x >> 2;        // 0..511
    const int bc = blockIdx.x & 3;         // 0..3
    const int wr = wave >> 2;              // 0..1
    const int wc = wave & 3;               // 0..3
    const int row0 = br * 128 + wr * 64;
    const int col0 = bc * 256 + wc * 64;
    const int lh = lane & 15;              // A: M-row / B,C: N-col
    const int kh = (lane >> 4) << 1;       // lanes 0-15: K+{0,1}; 16-31: K+{2,3}

    v8f acc[4][4] = {};
    const float* aBase = x    + (size_t)(row0 + lh) * I_ + kh;
    const float* bBase = W_in + (size_t)(col0 + lh) * I_ + kh;

    for (int k = 0; k < I_; k += 4) {
        v2f af[4], bf[4];
#pragma unroll
        for (int i = 0; i < 4; ++i)
            af[i] = *(const v2f*)(aBase + (size_t)(i * 16) * I_ + k);
#pragma unroll
        for (int j = 0; j < 4; ++j)
            bf[j] = *(const v2f*)(bBase + (size_t)(j * 16) * I_ + k);
#pragma unroll
        for (int i = 0; i < 4; ++i)
#pragma unroll
            for (int j = 0; j < 4; ++j)
                acc[i][j] = __builtin_amdgcn_wmma_f32_16x16x4_f32(
                    false, af[i], false, bf[j], (short)0, acc[i][j], false, false);
    }

    const int mbase = (lane >> 4) << 3;    // 0 or 8
#pragma unroll
    for (int j = 0; j < 4; ++j) {
        const int hcol = col0 + j * 16 + lh;
        const float bi = b_in[hcol];
#pragma unroll
        for (int i = 0; i < 4; ++i)
#pragma unroll
            for (int r = 0; r < 8; ++r) {
                const size_t row = (size_t)(row0 + i * 16 + mbase + r);
                ext[row * H_ + hcol] = acc[i][j][r] + bi;
            }
    }
}

// Serial-section step with precomputed ext:
//   h_out = relu( h_prev*(1-a) + ( ext_t + (h_prev@W_h^T + b_h)*gates ) * a )
// Grid: 32 blocks x 128 threads (4 waves). Wave tile: 32 rows x 64 cols
// = 2x4 WMMA tiles, single accumulator set (8 x v8f).
__global__ __launch_bounds__(128) void ctrnn_step_rec_kernel(
    const float* __restrict__ ext_t,   // B x H (includes b_in)
    const float* __restrict__ W_h,     // H x H
    const float* __restrict__ b_h,     // H
    const float* __restrict__ gates,   // B x H
    const float* __restrict__ h_prev,  // B x H
    float* __restrict__ h_out)         // B x H
{
    const int wave = threadIdx.x >> 5;      // 0..3
    const int lane = threadIdx.x & 31;
    const int job  = blockIdx.x * 4 + wave; // 0..127
    const int row0 = (job >> 4) * 32;       // batch rows
    const int col0 = (job & 15) * 64;       // hidden cols
    const int lh = lane & 15;
    const int kh = (lane >> 4) << 1;

    v8f acc[2][4] = {};
    const float* aBase = h_prev + (size_t)(row0 + lh) * H_ + kh;
    const float* bBase = W_h    + (size_t)(col0 + lh) * H_ + kh;

    for (int k = 0; k < H_; k += 4) {
        v2f af[2], bf[4];
#pragma unroll
        for (int i = 0; i < 2; ++i)
            af[i] = *(const v2f*)(aBase + (size_t)(i * 16) * H_ + k);
#pragma unroll
        for (int j = 0; j < 4; ++j)
            bf[j] = *(const v2f*)(bBase + (size_t)(j * 16) * H_ + k);
#pragma unroll
        for (int i = 0; i < 2; ++i)
#pragma unroll
            for (int j = 0; j < 4; ++j)
                acc[i][j] = __builtin_amdgcn_wmma_f32_16x16x4_f32(
                    false, af[i], false, bf[j], (short)0, acc[i][j], false, false);
    }

    const int mbase = (lane >> 4) << 3;
#pragma unroll
    for (int j = 0; j < 4; ++j) {
        const int hcol = col0 + j * 16 + lh;
        const float bh = b_h[hcol];
#pragma unroll
        for (int i = 0; i < 2; ++i)
#pragma unroll
            for (int r = 0; r < 8; ++r) {
                const int b = row0 + i * 16 + mbase + r;
                const size_t o = (size_t)b * H_ + hcol;
                const float g  = gates[o];
                const float hp = h_prev[o];
                const float e  = ext_t[o];
                const float rc = (acc[i][j][r] + bh) * g;
                const float hn = hp * (1.0f - ALPHA_) + (e + rc) * ALPHA_;
                h_out[o] = fmaxf(hn, 0.0f);
            }
    }
}

// Fallback fused step (small workspace): both GEMMs in one kernel.
// Grid: 32 blocks x 256 threads. Wave tile: 16 rows x 64 cols, two acc sets.
__global__ __launch_bounds__(256) void ctrnn_step_fused_kernel(
    int t,
    const float* __restrict__ x,       // T x B x I
    const float* __restrict__ W_in,    // H x I
    const float* __restrict__ b_in,    // H
    const float* __restrict__ W_h,     // H x H
    const float* __restrict__ b_h,     // H
    const float* __restrict__ gates,   // B x H
    const float* __restrict__ h_prev,  // B x H
    float* __restrict__ h_out)         // B x H
{
    const int wave = threadIdx.x >> 5;
    const int lane = threadIdx.x & 31;
    const int rowTile = blockIdx.x >> 1;
    const int colHalf = blockIdx.x & 1;
    const int cg   = colHalf * 8 + wave;
    const int row0 = rowTile * 16;
    const int col0 = cg * 64;
    const int lh = lane & 15;
    const int kh = (lane >> 4) << 1;

    const float* xt = x + (size_t)t * B_ * I_;

    v8f accI[4] = {};
    v8f accR[4] = {};

    const float* aBaseX = xt + (size_t)(row0 + lh) * I_ + kh;
    for (int k = 0; k < I_; k += 4) {
        v2f a = *(const v2f*)(aBaseX + k);
#pragma unroll
        for (int j = 0; j < 4; ++j) {
            v2f bfrag = *(const v2f*)(W_in + (size_t)(col0 + j * 16 + lh) * I_ + k + kh);
            accI[j] = __builtin_amdgcn_wmma_f32_16x16x4_f32(
                false, a, false, bfrag, (short)0, accI[j], false, false);
        }
    }

    const float* aBaseH = h_prev + (size_t)(row0 + lh) * H_ + kh;
    for (int k = 0; k < H_; k += 4) {
        v2f a = *(const v2f*)(aBaseH + k);
#pragma unroll
        for (int j = 0; j < 4; ++j) {
            v2f bfrag = *(const v2f*)(W_h + (size_t)(col0 + j * 16 + lh) * H_ + k + kh);
            accR[j] = __builtin_amdgcn_wmma_f32_16x16x4_f32(
                false, a, false, bfrag, (short)0, accR[j], false, false);
        }
    }

    const int mbase = (lane >> 4) << 3;
#pragma unroll
    for (int j = 0; j < 4; ++j) {
        const int hcol = col0 + j * 16 + lh;
        const float bi = b_in[hcol];
        const float bh = b_h[hcol];
#pragma unroll
        for (int r = 0; r < 8; ++r) {
            const int b = row0 + mbase + r;
            const size_t o = (size_t)b * H_ + hcol;
            const float g  = gates[o];
            const float hp = h_prev[o];
            const float e  = accI[j][r] + bi;
            const float rc = (accR[j][r] + bh) * g;
            const float hn = hp * (1.0f - ALPHA_) + (e + rc) * ALPHA_;
            h_out[o] = fmaxf(hn, 0.0f);
        }
    }
}

extern "C" void kernel_launch(void* const* d_in, const int* in_sizes, int n_in,
                              void* d_out, int out_size, void* d_ws, size_t ws_size,
                              hipStream_t stream) {
    (void)in_sizes; (void)n_in; (void)out_size;

    const float* x      = (const float*)d_in[0];
    const float* sub_id = (const float*)d_in[1];
    const float* W_in   = (const float*)d_in[2];
    const float* b_in   = (const float*)d_in[3];
    const float* W_h    = (const float*)d_in[4];
    const float* b_h    = (const float*)d_in[5];
    const float* mask   = (const float*)d_in[6];
    float* out = (float*)d_out;                 // [T*B*H] trajectory, then [B*H] h_last

    float* gates = (float*)d_ws;                // B*H floats
    float* h0    = gates + (size_t)B_ * H_;     // B*H floats (zeros)
    float* ext   = h0    + (size_t)B_ * H_;     // (T*B)*H floats, if it fits

    const size_t need = ((size_t)B_ * H_ * 2 + (size_t)T_ * B_ * H_) * sizeof(float);

    hipMemsetAsync(h0, 0, (size_t)B_ * H_ * sizeof(float), stream);
    ctrnn_gates_kernel<<<(B_ * H_) / 256, 256, 0, stream>>>(sub_id, mask, gates);

    if (ws_size >= need) {
        // Hoist the input projection out of the serial scan (whole-chip GEMM).
        ctrnn_ext_kernel<<<2048, 256, 0, stream>>>(x, W_in, b_in, ext);
        for (int t = 0; t < T_; ++t) {
            const float* hp = (t == 0) ? h0 : out + (size_t)(t - 1) * B_ * H_;
            float* ho = out + (size_t)t * B_ * H_;
            ctrnn_step_rec_kernel<<<32, 128, 0, stream>>>(
                ext + (size_t)t * B_ * H_, W_h, b_h, gates, hp, ho);
        }
    } else {
        for (int t = 0; t < T_; ++t) {
            const float* hp = (t == 0) ? h0 : out + (size_t)(t - 1) * B_ * H_;
            float* ho = out + (size_t)t * B_ * H_;
            ctrnn_step_fused_kernel<<<32, 256, 0, stream>>>(
                t, x, W_in, b_in, W_h, b_h, gates, hp, ho);
        }
    }

    // h_last = out[T-1]
    hipMemcpyAsync(out + (size_t)T_ * B_ * H_, out + (size_t)(T_ - 1) * B_ * H_,
                   (size_t)B_ * H_ * sizeof(float), hipMemcpyDeviceToDevice, stream);
}